// GCNEncoder_28286654612005
// MI455X (gfx1250) — compile-verified
//
#include <hip/hip_runtime.h>
#include <stdint.h>

typedef __attribute__((ext_vector_type(2))) float v2f;
typedef __attribute__((ext_vector_type(8))) float v8f;

#define DIM_IN  128
#define DIM_HID 64
#define DIM_EMB 32

__device__ __forceinline__ void atomic_add_f32(float* p, float v) {
  // relaxed agent-scope fp add on coarse-grained memory -> global_atomic_add_f32
  __hip_atomic_fetch_add(p, v, __ATOMIC_RELAXED, __HIP_MEMORY_SCOPE_AGENT);
}

// ---------------- degree / normalization ----------------
__global__ void k_deg_init(float* __restrict__ deg, int N) {
  int i = blockIdx.x * blockDim.x + threadIdx.x;
  if (i < N) deg[i] = 1.0f;  // self-loop contribution
}

__global__ void k_deg_count(const int64_t* __restrict__ dst, float* __restrict__ deg, int E) {
  int e = blockIdx.x * blockDim.x + threadIdx.x;
  if (e < E) atomic_add_f32(&deg[(int)dst[e]], 1.0f);
}

__global__ void k_dinv(float* __restrict__ deg, int N) {
  int i = blockIdx.x * blockDim.x + threadIdx.x;
  if (i < N) deg[i] = rsqrtf(deg[i]);  // deg >= 1 always (self-loops)
}

// ---------------- fp32 WMMA GEMM: Out[M x 64] = A[M x K] @ W ----------------
// One wave computes a full 16x64 output strip: A loaded once per K-step,
// 4 independent WMMAs (one per 16-col tile) -> 4 accumulators.
// Columns [0,split) from WL, columns [split,64) from WR; both row-major
// with compile-time stride WSTR.
template <int K, int WSTR>
__global__ __launch_bounds__(256)
void k_gemm_wmma(const float* __restrict__ A, int M,
                 const float* __restrict__ WL, const float* __restrict__ WR,
                 int split, float* __restrict__ Out) {
  const int lane = threadIdx.x & 31;
  const int wave = threadIdx.x >> 5;
  const int numM = (M + 15) >> 4;
  const int mtile = blockIdx.x * 8 + wave;
  if (mtile >= numM) return;  // whole-wave guard: EXEC all-ones for WMMA

  const int lid  = lane & 15;
  const int half = lane >> 4;          // K split across lane halves
  const int mbase = mtile * 16;

  int row = mbase + lid;
  if (row >= M) row = M - 1;           // clamp tail loads (stores guarded)
  const float* __restrict__ ah = A + (size_t)row * K + half * 2;

  // per-lane weight base pointers, one per 16-col tile; loop offsets become
  // compile-time immediates (kb*WSTR*4 <= 32KB < 24-bit ioffset)
  const float* wh[4];
#pragma unroll
  for (int t = 0; t < 4; ++t) {
    const int col = t * 16 + lid;
    const float* Wp = (col < split) ? (WL + col) : (WR + (col - split));
    wh[t] = Wp + half * 2 * WSTR;
  }

  v8f acc0 = {}, acc1 = {}, acc2 = {}, acc3 = {};
#pragma unroll
  for (int kb = 0; kb < K; kb += 4) {
    v2f a = *(const v2f*)(ah + kb);    // 8B-aligned (kb, half*2 even)
    v2f b0, b1, b2, b3;
    b0.x = wh[0][kb * WSTR]; b0.y = wh[0][(kb + 1) * WSTR];
    b1.x = wh[1][kb * WSTR]; b1.y = wh[1][(kb + 1) * WSTR];
    b2.x = wh[2][kb * WSTR]; b2.y = wh[2][(kb + 1) * WSTR];
    b3.x = wh[3][kb * WSTR]; b3.y = wh[3][(kb + 1) * WSTR];
    acc0 = __builtin_amdgcn_wmma_f32_16x16x4_f32(false, a, false, b0, (short)0, acc0, false, false);
    acc1 = __builtin_amdgcn_wmma_f32_16x16x4_f32(false, a, false, b1, (short)0, acc1, false, false);
    acc2 = __builtin_amdgcn_wmma_f32_16x16x4_f32(false, a, false, b2, (short)0, acc2, false, false);
    acc3 = __builtin_amdgcn_wmma_f32_16x16x4_f32(false, a, false, b3, (short)0, acc3, false, false);
  }

  // D layout: VGPR r -> M = mbase + half*8 + r, N = tile*16 + lid
  float* __restrict__ ob = Out + (size_t)(mbase + half * 8) * DIM_HID + lid;
#pragma unroll
  for (int r = 0; r < 8; ++r) {
    if (mbase + half * 8 + r < M) {
      ob[(size_t)r * DIM_HID +  0] = acc0[r];
      ob[(size_t)r * DIM_HID + 16] = acc1[r];
      ob[(size_t)r * DIM_HID + 32] = acc2[r];
      ob[(size_t)r * DIM_HID + 48] = acc3[r];
    }
  }
}

// ---------------- propagation ----------------
// Acc[i][c] = dinv[i]^2 * Hin[i][c]   (self-loop term, initializes Acc)
__global__ void k_selfloop(const float* __restrict__ dinv, const float* __restrict__ Hin,
                           float* __restrict__ Acc, int total) {
  int idx = blockIdx.x * blockDim.x + threadIdx.x;
  if (idx < total) {
    float w = dinv[idx >> 6];
    Acc[idx] = w * w * Hin[idx];
  }
}

// Acc[dst[e]][c] += dinv[src]*dinv[dst] * Hin[src[e]][c]
// One thread per (edge, column); e is uniform across each wave32 -> index and
// dinv loads collapse to one cacheline, gather/atomic fully coalesced.
__global__ void k_scatter(const int64_t* __restrict__ src, const int64_t* __restrict__ dst,
                          const float* __restrict__ dinv, const float* __restrict__ Hin,
                          float* __restrict__ Acc, long long total) {
  long long idx = (long long)blockIdx.x * blockDim.x + threadIdx.x;
  if (idx < total) {
    const int e = (int)(idx >> 6);
    const int c = (int)(idx & 63);
    const int s = (int)src[e];
    const int d = (int)dst[e];
    const float w = dinv[s] * dinv[d];
    atomic_add_f32(&Acc[(size_t)d * 64 + c], w * Hin[(size_t)s * 64 + c]);
  }
}

__global__ void k_bias_relu(float* __restrict__ H, const float* __restrict__ b, int total) {
  int idx = blockIdx.x * blockDim.x + threadIdx.x;
  if (idx < total) H[idx] = fmaxf(H[idx] + b[idx & 63], 0.0f);
}

// split 64-wide accumulator into mu (cols 0-31) and logvar (cols 32-63) + bias
__global__ void k_final(const float* __restrict__ Acc, const float* __restrict__ bmu,
                        const float* __restrict__ blv, float* __restrict__ out,
                        int N, int total) {
  int idx = blockIdx.x * blockDim.x + threadIdx.x;
  if (idx < total) {
    const int i = idx >> 6;
    const int c = idx & 63;
    if (c < 32) out[(size_t)i * 32 + c] = Acc[idx] + bmu[c];
    else out[(size_t)N * 32 + (size_t)i * 32 + (c - 32)] = Acc[idx] + blv[c - 32];
  }
}

// ---------------- driver ----------------
extern "C" void kernel_launch(void* const* d_in, const int* in_sizes, int n_in,
                              void* d_out, int out_size, void* d_ws, size_t ws_size,
                              hipStream_t stream) {
  const float*   x     = (const float*)d_in[0];
  const int64_t* edges = (const int64_t*)d_in[1];   // [2, E] int64
  const float*   W1    = (const float*)d_in[2];     // [128, 64] row-major
  const float*   b1    = (const float*)d_in[3];
  const float*   Wmu   = (const float*)d_in[4];     // [64, 32]
  const float*   bmu   = (const float*)d_in[5];
  const float*   Wlv   = (const float*)d_in[6];     // [64, 32]
  const float*   blv   = (const float*)d_in[7];

  const int N = in_sizes[0] / DIM_IN;
  const int E = in_sizes[1] / 2;
  const int64_t* src = edges;
  const int64_t* dst = edges + E;

  float* dinv = (float*)d_ws;              // N        (deg -> dinv in place)
  float* bufA = dinv + N;                  // N*64     (h0, then hc)
  float* bufB = bufA + (size_t)N * 64;     // N*64     (agg1 -> h -> agg2)

  const int NT = 256;
  dim3 blk(NT);
  const int totalNC = N * 64;
  const long long totalEC = (long long)E * 64;
  const unsigned gN  = (unsigned)((N + NT - 1) / NT);
  const unsigned gE  = (unsigned)((E + NT - 1) / NT);
  const unsigned gNC = (unsigned)((totalNC + NT - 1) / NT);
  const unsigned gEC = (unsigned)((totalEC + NT - 1) / NT);

  // normalization coefficients
  k_deg_init<<<gN, blk, 0, stream>>>(dinv, N);
  k_deg_count<<<gE, blk, 0, stream>>>(dst, dinv, E);
  k_dinv<<<gN, blk, 0, stream>>>(dinv, N);

  const int numM = (N + 15) / 16;
  dim3 gGemm((numM + 7) / 8);

  // ---- layer 1: h = relu(propagate(x @ W1) + b1) ----
  k_gemm_wmma<DIM_IN, DIM_HID><<<gGemm, blk, 0, stream>>>(x, N, W1, W1, DIM_HID, bufA);
  k_selfloop<<<gNC, blk, 0, stream>>>(dinv, bufA, bufB, totalNC);
  k_scatter<<<gEC, blk, 0, stream>>>(src, dst, dinv, bufA, bufB, totalEC);
  k_bias_relu<<<gNC, blk, 0, stream>>>(bufB, b1, totalNC);

  // ---- layer 2: mu / logvar share one fused GEMM + one propagation ----
  k_gemm_wmma<DIM_HID, DIM_EMB><<<gGemm, blk, 0, stream>>>(bufB, N, Wmu, Wlv, DIM_EMB, bufA);
  k_selfloop<<<gNC, blk, 0, stream>>>(dinv, bufA, bufB, totalNC);
  k_scatter<<<gEC, blk, 0, stream>>>(src, dst, dinv, bufA, bufB, totalEC);
  k_final<<<gNC, blk, 0, stream>>>(bufB, bmu, blv, (float*)d_out, N, totalNC);
}